// GraphAttentionLayer_678604833046
// MI455X (gfx1250) — compile-verified
//
#include <hip/hip_runtime.h>
#include <hip/hip_bf16.h>

typedef __attribute__((ext_vector_type(2))) float v2f;
typedef __attribute__((ext_vector_type(8))) float v8f;

#define B_ 4
#define N_ 64
#define T_ 64
#define FH_ 128
#define FO_ 64

// -------------------------------------------------------------------------
// Kernel 1: Wh[b,n,t,o] = sum_k h[b,n,t,k] * W[o,k] + bias[o]
// M = B*N*T = 16384 rows, K = 128, O = 64.  One wave -> one 16x16 tile.
// f32 WMMA, K stepped by 4.
// -------------------------------------------------------------------------
__global__ void wh_gemm_kernel(const float* __restrict__ h,
                               const float* __restrict__ W,
                               const float* __restrict__ bias,
                               float* __restrict__ Wh) {
  const int wave = blockIdx.x * (blockDim.x >> 5) + (threadIdx.x >> 5);
  const int lane = threadIdx.x & 31;
  const int otile = wave & 3;        // 4 tiles over O=64
  const int mtile = wave >> 2;       // 1024 tiles over M=16384
  const int ln = lane & 15;          // A: row m; B: col n (both = lane%16)
  const int kh = lane >> 4;          // lane-half selects K pair {0,1} vs {2,3}

  const float* arow = h + (size_t)(mtile * 16 + ln) * FH_;
  const float* brow = W + (size_t)(otile * 16 + ln) * FH_;

  v8f c = {0.f, 0.f, 0.f, 0.f, 0.f, 0.f, 0.f, 0.f};
#pragma unroll 4
  for (int k0 = 0; k0 < FH_; k0 += 4) {
    const int k = k0 + 2 * kh;
    v2f a = *(const v2f*)(arow + k);
    v2f b = *(const v2f*)(brow + k);
    c = __builtin_amdgcn_wmma_f32_16x16x4_f32(false, a, false, b,
                                              (short)0, c, false, false);
  }

  const float bo = bias[otile * 16 + ln];
  const int mrow = mtile * 16 + 8 * kh;          // C/D: lane-half adds 8 to M
  float* out = Wh + (size_t)mrow * FO_ + otile * 16 + ln;
#pragma unroll
  for (int v = 0; v < 8; ++v) out[(size_t)v * FO_] = c[v] + bo;
}

// -------------------------------------------------------------------------
// Kernel 2: pr_t[b,t,n] = <Wh[b,n,t,:], a_r>, ps_t[b,t,n] = <Wh[b,n,t,:], a_s>
// One thread per (b,n,t); float4 vector loads.
// -------------------------------------------------------------------------
__global__ void node_proj_kernel(const float* __restrict__ Wh,
                                 const float* __restrict__ aw,
                                 float* __restrict__ pr_t,
                                 float* __restrict__ ps_t) {
  const int idx = blockIdx.x * blockDim.x + threadIdx.x;   // (b*N+n)*T + t
  const int t = idx & (T_ - 1);
  const int bn = idx >> 6;
  const int n = bn & (N_ - 1);
  const int b = bn >> 6;

  const float4* row = (const float4*)(Wh + (size_t)idx * FO_);
  const float4* ar4 = (const float4*)(aw);
  const float4* as4 = (const float4*)(aw + FO_);
  float pr = 0.f, ps = 0.f;
#pragma unroll
  for (int i = 0; i < FO_ / 4; ++i) {
    const float4 w = row[i];
    const float4 r = ar4[i];
    const float4 s = as4[i];
    pr += w.x * r.x + w.y * r.y + w.z * r.z + w.w * r.w;
    ps += w.x * s.x + w.y * s.y + w.z * s.z + w.w * s.w;
  }
  const int o = (b * T_ + t) * N_ + n;
  pr_t[o] = pr;
  ps_t[o] = ps;
}

// -------------------------------------------------------------------------
// Kernel 3: per (b,t) build A (64x64), row sums, then A_sym and A_norm.
// A[b,t,r,s] = exp(leaky_relu(pr[r] + ps[s] + ab)) for r!=s, 0 on diagonal.
// LDS tile with stride 65 to avoid bank conflicts on the transposed read.
// -------------------------------------------------------------------------
__global__ void adjacency_kernel(const float* __restrict__ pr_t,
                                 const float* __restrict__ ps_t,
                                 const float* __restrict__ ab_p,
                                 float* __restrict__ A,
                                 float* __restrict__ A_sym,
                                 float* __restrict__ A_norm) {
  __shared__ float tile[N_ * 65];
  __shared__ float rs[N_];
  __shared__ float prs[N_];
  __shared__ float pss[N_];

  const int bt = blockIdx.x;          // b*T + t
  const int tid = threadIdx.x;        // 0..255
  const float ab = ab_p[0];

  if (tid < N_) {
    prs[tid] = pr_t[bt * N_ + tid];
    pss[tid] = ps_t[bt * N_ + tid];
  }
  __syncthreads();

  const int s = tid & (N_ - 1);
  const int r0 = tid >> 6;            // 0..3
  const size_t base = (size_t)bt * (N_ * N_);

#pragma unroll
  for (int rr = 0; rr < 16; ++rr) {
    const int r = r0 * 16 + rr;
    float val = 0.f;
    if (r != s) {
      float x = prs[r] + pss[s] + ab;
      x = (x > 0.f) ? x : 0.2f * x;   // leaky_relu slope 0.2
      val = expf(x);
    }
    tile[r * 65 + s] = val;
    A[base + r * N_ + s] = val;
  }
  __syncthreads();

  if (tid < N_) {
    float sum = 0.f;
#pragma unroll 8
    for (int ss = 0; ss < N_; ++ss) sum += tile[tid * 65 + ss];
    rs[tid] = sum;
  }
  __syncthreads();

#pragma unroll
  for (int rr = 0; rr < 16; ++rr) {
    const int r = r0 * 16 + rr;
    const float v = tile[r * 65 + s];
    A_sym[base + r * N_ + s] = 0.5f * (v + tile[s * 65 + r]);
    A_norm[base + r * N_ + s] = v / rs[r];
  }
}

// -------------------------------------------------------------------------
// Kernel 4: h_p[b,n,t,o] = relu( sum_s A_norm[b,t,n,s] * Wh[b,s,t,o] )
// 256 independent 64x64x64 fp32 GEMMs; one wave -> one 16x16 tile, K step 4.
// -------------------------------------------------------------------------
__global__ void hp_gemm_kernel(const float* __restrict__ A_norm,
                               const float* __restrict__ Wh,
                               float* __restrict__ h_p) {
  const int wave = blockIdx.x * (blockDim.x >> 5) + (threadIdx.x >> 5);
  const int lane = threadIdx.x & 31;
  const int tileid = wave & 15;       // 16 tiles per (b,t)
  const int bt = wave >> 4;           // 0..255
  const int b = bt >> 6;
  const int t = bt & (T_ - 1);
  const int otile = tileid & 3;
  const int mtile = tileid >> 2;
  const int ln = lane & 15;
  const int kh = lane >> 4;

  const float* arow = A_norm + (size_t)bt * (N_ * N_) + (size_t)(mtile * 16 + ln) * N_;
  // B[k][n] = Wh[b, k, t, otile*16+n]; stride over k is T_*FO_ floats
  const float* bbase = Wh + (((size_t)b * N_) * T_ + t) * FO_ + otile * 16 + ln;

  v8f c = {0.f, 0.f, 0.f, 0.f, 0.f, 0.f, 0.f, 0.f};
#pragma unroll 4
  for (int k0 = 0; k0 < N_; k0 += 4) {
    const int k = k0 + 2 * kh;
    v2f a = *(const v2f*)(arow + k);
    v2f bf;
    bf.x = bbase[(size_t)k * (T_ * FO_)];
    bf.y = bbase[(size_t)(k + 1) * (T_ * FO_)];
    c = __builtin_amdgcn_wmma_f32_16x16x4_f32(false, a, false, bf,
                                              (short)0, c, false, false);
  }

  const int mrow = mtile * 16 + 8 * kh;
#pragma unroll
  for (int v = 0; v < 8; ++v) {
    float x = c[v];
    x = (x > 0.f) ? x : 0.f;
    h_p[(((size_t)(b * N_ + mrow + v) * T_) + t) * FO_ + otile * 16 + ln] = x;
  }
}

// -------------------------------------------------------------------------
// Host launcher
// inputs: 0:h 1:rel_rec 2:rel_send 3:W_fc 4:b_fc 5:a_w 6:a_b
// d_out:  [h_p | A | A_sym | A_norm], each B*T*N*N (= B*N*T*FO) = 1048576 f32
// d_ws:   [Wh: 4 MB][pr_t: 64 KB][ps_t: 64 KB]
// -------------------------------------------------------------------------
extern "C" void kernel_launch(void* const* d_in, const int* in_sizes, int n_in,
                              void* d_out, int out_size, void* d_ws, size_t ws_size,
                              hipStream_t stream) {
  const float* h    = (const float*)d_in[0];
  const float* W_fc = (const float*)d_in[3];
  const float* b_fc = (const float*)d_in[4];
  const float* a_w  = (const float*)d_in[5];
  const float* a_b  = (const float*)d_in[6];

  const size_t nWh = (size_t)B_ * N_ * T_ * FO_;     // 1048576
  float* Wh   = (float*)d_ws;
  float* pr_t = Wh + nWh;
  float* ps_t = pr_t + (size_t)B_ * T_ * N_;

  float* out    = (float*)d_out;
  float* h_p    = out;
  float* A      = out + nWh;
  float* A_sym  = out + 2 * nWh;
  float* A_norm = out + 3 * nWh;

  // Kernel 1: 4096 tile-waves (1024 M-tiles x 4 O-tiles), 8 waves/block.
  wh_gemm_kernel<<<512, 256, 0, stream>>>(h, W_fc, b_fc, Wh);

  // Kernel 2: one thread per (b,n,t) = 16384 threads.
  node_proj_kernel<<<64, 256, 0, stream>>>(Wh, a_w, pr_t, ps_t);

  // Kernel 3: one block per (b,t) = 256 blocks.
  adjacency_kernel<<<B_ * T_, 256, 0, stream>>>(pr_t, ps_t, a_b, A, A_sym, A_norm);

  // Kernel 4: 256 (b,t) GEMMs x 16 tiles = 4096 waves, 8 waves/block.
  hp_gemm_kernel<<<512, 256, 0, stream>>>(A_norm, Wh, h_p);
}